// DifferentiableTree_42314017800718
// MI455X (gfx1250) — compile-verified
//
#include <hip/hip_runtime.h>

// ---------------------------------------------------------------------------
// Batched forward kinematics on gfx1250 (MI455X).
//
// Roofline: per batch element 32B in + 576B out => HBM-bound (~6.8us floor at
// 23.3 TB/s). Chain step T_next = T_p * (c*M0 + s*M1 + (1-c)*M2 + M3) with
// shared per-link M matrices; implemented as V_WMMA_F32_16X16X4_F32 with
//   A (16x4, shared, in regs) = [M0^T; M1^T; M2^T; M3^T]
//   B (4x16)                  = 4 batch states S_b = T_b^T side by side
// D(4j+r, 4b+c) = (T_b * Mj)(c, r). Recombine with Rodrigues scalars needs
// only a half-wave swap (ds_swizzle SWAPX16), result is again B-layout.
// Chunk index is forced wave-uniform (readfirstlane) so the guard is a scalar
// branch and every WMMA executes with EXEC == all-1s, as the ISA requires.
// Output is streamed with non-temporal stores (write-once, never re-read).
// ---------------------------------------------------------------------------

typedef float v2f __attribute__((ext_vector_type(2)));
typedef float v8f __attribute__((ext_vector_type(8)));

#define N_LINKS 8
#define CPW 4 // chunks (of 4 batch elements) per wave

__device__ __forceinline__ float swap_half(float x) {
  // lanes 0-15 <-> 16-31 : group-of-32 swizzle, xor_mask=0x10, and_mask=0x1f
  return __int_as_float(__builtin_amdgcn_ds_swizzle(__float_as_int(x), 0x401f));
}

__global__ __launch_bounds__(256) void fk_wmma_kernel(
    const float* __restrict__ q,    // [B,8]
    const float* __restrict__ fr,   // [8,3,3] fixed_rot
    const float* __restrict__ ft,   // [8,3]   fixed_trans
    const float* __restrict__ ax,   // [8,3]   joint_axis (unit)
    float* __restrict__ out,        // [B,9,4,4]
    int nChunks)
{
  const int lane = threadIdx.x & 31;
  // Wave index: provably uniform -> scalar chunk guard, full EXEC at WMMA.
  const int gwave = __builtin_amdgcn_readfirstlane(
      blockIdx.x * (blockDim.x >> 5) + (threadIdx.x >> 5));

  const int n   = lane & 15;        // column index within 16 (N dim)
  const int rr  = n & 3;            // index within 4-block
  const int jb  = n >> 2;           // block 0..3: matrix j (for A) / batch (for B,D)
  const int kb  = (lane >> 4) << 1; // K rows held by this half-wave: {kb, kb+1}

  // ---- Precompute shared A = [M0^T;M1^T;M2^T;M3^T] per link, kept in VGPRs.
  // A(m=4j+r, k) = Mj(k, r); lane holds (m = n, k = kb) and (m = n, k = kb+1).
  float Aa[N_LINKS], Ab[N_LINKS];
  {
    const int rc = (rr < 3) ? rr : 2; // clamped column for safe loads
#pragma unroll
    for (int i = 0; i < N_LINKS; ++i) {
      const float a0 = ax[i * 3 + 0], a1 = ax[i * 3 + 1], a2 = ax[i * 3 + 2];
      // skew(a) column rc
      float kc0, kc1, kc2;
      if (rc == 0)      { kc0 = 0.f;  kc1 = a2;  kc2 = -a1; }
      else if (rc == 1) { kc0 = -a2;  kc1 = 0.f; kc2 = a0;  }
      else              { kc0 = a1;   kc1 = -a0; kc2 = 0.f; }
      const float arr = (rc == 0) ? a0 : ((rc == 1) ? a1 : a2);
#pragma unroll
      for (int t = 0; t < 2; ++t) {
        const int k   = kb + t;
        const int kcl = (k < 3) ? k : 2; // clamped row for safe loads
        const float r0 = fr[i * 9 + kcl * 3 + 0];
        const float r1 = fr[i * 9 + kcl * 3 + 1];
        const float r2 = fr[i * 9 + kcl * 3 + 2];
        const float m0 = (rc == 0) ? r0 : ((rc == 1) ? r1 : r2);   // Rf(k,rr)
        const float m1 = r0 * kc0 + r1 * kc1 + r2 * kc2;           // (Rf*K)(k,rr)
        const float m2 = (r0 * a0 + r1 * a1 + r2 * a2) * arr;      // (Rf*aa^T)(k,rr)
        const float m3 = (rr == 3) ? ((k < 3) ? ft[i * 3 + kcl] : 1.f) : 0.f;
        float v;
        if (jb == 3) {
          v = m3;
        } else {
          v = (jb == 0) ? m0 : ((jb == 1) ? m1 : m2);
          if (k >= 3 || rr >= 3) v = 0.f; // rotation blocks are 3x3
        }
        if (t == 0) Aa[i] = v; else Ab[i] = v;
      }
    }
  }

  const bool laneLow = (lane < 16);

  for (int c = 0; c < CPW; ++c) {
    const int chunk = gwave * CPW + c;    // uniform (SGPR)
    if (chunk >= nChunks) continue;       // scalar branch: EXEC stays full
    const int b = chunk * 4 + jb;         // this lane's batch element

    // q[b][0..8): two coalesced b128 loads
    const float4 q0 = *(const float4*)(q + (size_t)b * 8);
    const float4 q1 = *(const float4*)(q + (size_t)b * 8 + 4);
    const float qarr[8] = {q0.x, q0.y, q0.z, q0.w, q1.x, q1.y, q1.z, q1.w};

    // State S = T^T in B layout: lane holds S(kb, rr)=T(rr,kb), S(kb+1, rr).
    float s0 = (rr == kb)     ? 1.f : 0.f;
    float s1 = (rr == kb + 1) ? 1.f : 0.f;

    // element address of T[b][l][rr][kb]; pair (kb, kb+1) is contiguous -> b64
    float* obase = out + (size_t)b * 144 + rr * 4 + kb;
    { v2f w; w.x = s0; w.y = s1;                 // l = 0: identity
      __builtin_nontemporal_store(w, (v2f*)obase); }

#pragma unroll
    for (int i = 0; i < N_LINKS; ++i) {
      const float ci = __cosf(qarr[i]);
      const float si = __sinf(qarr[i]);
      const float oc = 1.f - ci;

      v2f A; A.x = Aa[i]; A.y = Ab[i];
      v2f B; B.x = s0;    B.y = s1;
      v8f C = {};
      // D = A x B : all four (T_b * Mj)^T products for 4 batch elements
      v8f D = __builtin_amdgcn_wmma_f32_16x16x4_f32(
          /*neg_a=*/false, A, /*neg_b=*/false, B,
          /*c_mod=*/(short)0, C, /*reuse_a=*/false, /*reuse_b=*/false);

      // VGPR r: lanes 0-15 = D(r, n), lanes 16-31 = D(r+8, n).
      // S_next(r, n) = ci*D(r,n) + si*D(r+4,n) + oc*D(r+8,n) + D(r+12,n)
      float o0, o1, o2, o3;
#pragma unroll
      for (int r = 0; r < 4; ++r) {
        const float dr   = D[r];
        const float dr4  = D[r + 4];
        const float sw   = swap_half(dr);   // lane n -> D(r+8,  n)
        const float sw4  = swap_half(dr4);  // lane n -> D(r+12, n)
        const float v = ci * dr + si * dr4 + oc * sw + sw4;
        if (r == 0) o0 = v; else if (r == 1) o1 = v;
        else if (r == 2) o2 = v; else o3 = v;
      }
      // Repack into B layout: vgpr0 = rows {0,2}, vgpr1 = rows {1,3}
      const float p2 = swap_half(o2);
      const float p3 = swap_half(o3);
      s0 = laneLow ? o0 : p2;
      s1 = laneLow ? o1 : p3;

      { v2f w; w.x = s0; w.y = s1;
        __builtin_nontemporal_store(w, (v2f*)(obase + (i + 1) * 16)); }
    }
  }
}

extern "C" void kernel_launch(void* const* d_in, const int* in_sizes, int n_in,
                              void* d_out, int out_size, void* d_ws, size_t ws_size,
                              hipStream_t stream) {
  (void)n_in; (void)out_size; (void)d_ws; (void)ws_size;
  const float* q  = (const float*)d_in[0];
  const float* fr = (const float*)d_in[1];
  const float* ft = (const float*)d_in[2];
  const float* ax = (const float*)d_in[3];
  float* out = (float*)d_out;

  const int B = in_sizes[0] / N_LINKS;       // 262144
  const int nChunks = B / 4;                 // 4 batch elements per WMMA chain
  const int wavesNeeded = (nChunks + CPW - 1) / CPW;
  const int blocks = (wavesNeeded + 7) / 8;  // 8 waves (256 threads) per block

  fk_wmma_kernel<<<blocks, 256, 0, stream>>>(q, fr, ft, ax, out, nChunks);
}